// Net_4861902979707
// MI455X (gfx1250) — compile-verified
//
#include <hip/hip_runtime.h>

// ---------------------------------------------------------------------------
// Pairwise-distance + tiny MLP (3 -> 2 -> 2 -> 1, ELU) for B = 4,194,304.
// Memory-bound (~168 MB @ 23.3 TB/s => ~7.2 us floor). Strategy:
//   * CDNA5 async copy engine: global_load_async_to_lds_b128 stages each
//     block's 9216 B tile into LDS (no VGPR bounce), s_wait_asynccnt + barrier
//   * per-lane distances in VALU
//   * layer 1 (3->2 GEMM) on the matrix pipe: V_WMMA_F32_16X16X4_F32,
//     K padded 3->4, N padded 2->16; two WMMAs per wave (16 samples each)
//   * ds_bpermute redistribution, ELU + layers 2/3 in VALU
//   * coalesced NT store
// ---------------------------------------------------------------------------

typedef float v2f __attribute__((ext_vector_type(2)));
typedef float v4f __attribute__((ext_vector_type(4)));
typedef float v8f __attribute__((ext_vector_type(8)));

#if defined(__gfx1250__) && __has_builtin(__builtin_amdgcn_wmma_f32_16x16x4_f32)
#define USE_WMMA 1
#else
#define USE_WMMA 0
#if defined(__HIP_DEVICE_COMPILE__)
#warning "CDNA5: __builtin_amdgcn_wmma_f32_16x16x4_f32 unavailable; VALU fallback for layer 1"
#endif
#endif

static __device__ __forceinline__ float eluf(float x) {
    // jax.nn.elu, alpha = 1
    return x > 0.0f ? x : (__expf(x) - 1.0f);
}

#define TPB 256
#define FLOATS_PER_TILE (TPB * 9)              // 2304 floats = 9216 bytes
#define VEC4_PER_TILE   (FLOATS_PER_TILE / 4)  // 576

__global__ __launch_bounds__(TPB) void pairdist_mlp_kernel(
    const float* __restrict__ X,
    const float* __restrict__ W1, const float* __restrict__ b1,
    const float* __restrict__ W2, const float* __restrict__ b2,
    const float* __restrict__ W3, const float* __restrict__ b3,
    float* __restrict__ out)
{
    __shared__ alignas(16) float sX[FLOATS_PER_TILE];

    const int t = threadIdx.x;
    const size_t tileBase = (size_t)blockIdx.x * FLOATS_PER_TILE; // float idx; 9216*blockIdx bytes

    // ---- Phase 1: async global -> LDS staging (CDNA5 ASYNCcnt path) -------
    // global addr = SGPR64(X) + per-lane i32 offset; LDS addr = low 32 bits of
    // the generic shared pointer (hardware adds the wave's LDS_BASE).
    {
        const unsigned ldsBase = (unsigned)(size_t)(&sX[0]);
        const int gByte = (int)(tileBase * sizeof(float));   // < 2^31 for this shape
        const int o0 = gByte + t * 16;
        asm volatile("global_load_async_to_lds_b128 %0, %1, %2"
                     :: "v"(ldsBase + (unsigned)(t * 16)), "v"(o0), "s"(X)
                     : "memory");
        asm volatile("global_load_async_to_lds_b128 %0, %1, %2"
                     :: "v"(ldsBase + (unsigned)(t * 16 + 4096)), "v"(o0 + 4096), "s"(X)
                     : "memory");
        if (t < (VEC4_PER_TILE - 512)) {   // last 64 vec4s of the 576-vec4 tile
            asm volatile("global_load_async_to_lds_b128 %0, %1, %2"
                         :: "v"(ldsBase + (unsigned)(t * 16 + 8192)), "v"(o0 + 8192), "s"(X)
                         : "memory");
        }
        asm volatile("s_wait_asynccnt 0" ::: "memory");
    }
    __syncthreads();

    // ---- Phase 2: per-lane pairwise distances -----------------------------
    const float* p = &sX[t * 9];
    const float x0 = p[0], y0 = p[1], z0 = p[2];
    const float x1 = p[3], y1 = p[4], z1 = p[5];
    const float x2 = p[6], y2 = p[7], z2 = p[8];

    float dx, dy, dz;
    dx = x0 - x1; dy = y0 - y1; dz = z0 - z1;
    const float d0 = sqrtf(dx * dx + dy * dy + dz * dz);   // pair (0,1)
    dx = x0 - x2; dy = y0 - y2; dz = z0 - z2;
    const float d1 = sqrtf(dx * dx + dy * dy + dz * dz);   // pair (0,2)
    dx = x1 - x2; dy = y1 - y2; dz = z1 - z2;
    const float d2 = sqrtf(dx * dx + dy * dy + dz * dz);   // pair (1,2)

    // ---- Weights (wave-uniform -> scalar loads) ---------------------------
    const float w1_00 = W1[0], w1_01 = W1[1], w1_02 = W1[2];   // W1[0,:]
    const float w1_10 = W1[3], w1_11 = W1[4], w1_12 = W1[5];   // W1[1,:]
    const float bb1_0 = b1[0], bb1_1 = b1[1];
    const float w2_00 = W2[0], w2_01 = W2[1];                  // W2[0,:]
    const float w2_10 = W2[2], w2_11 = W2[3];                  // W2[1,:]
    const float bb2_0 = b2[0], bb2_1 = b2[1];
    const float w3_0 = W3[0], w3_1 = W3[1];
    const float bb3 = b3[0];

    float h0, h1;   // layer-1 pre-activations for this lane's sample

#if USE_WMMA
    // ---- Phase 3: layer 1 as D = A(16x4) * B(4x16) + C on the matrix pipe
    const int  lane = t & 31;
    const bool hiHalf = lane >= 16;
    const int  n = lane & 15;          // output-feature column for B/C frags

    // B fragment (4x16): col n = (W1[n][0], W1[n][1], W1[n][2], 0), n<2 else 0
    // f32 layout: lanes 0-15 hold K=0 (v0), K=1 (v1); lanes 16-31 K=2 (v0), K=3 (v1)
    float Bv0 = (n == 0) ? (hiHalf ? w1_02 : w1_00)
              : (n == 1) ? (hiHalf ? w1_12 : w1_10) : 0.0f;
    float Bv1 = hiHalf ? 0.0f
              : ((n == 0) ? w1_01 : (n == 1) ? w1_11 : 0.0f);
    v2f Bm; Bm[0] = Bv0; Bm[1] = Bv1;

    // C fragment: bias broadcast over rows (same value in all 8 acc regs)
    const float cval = (n == 0) ? bb1_0 : (n == 1) ? bb1_1 : 0.0f;
    v8f C = {cval, cval, cval, cval, cval, cval, cval, cval};

    // A fragment #1: rows = samples of lanes 0..15
    // lanes 0-15: (K0,K1) = (d0,d1) of own sample; lanes 16-31: (K2,K3) = (d2 of lane-16, 0)
    const float d2lo = __shfl(d2, n, 32);
    v2f A1; A1[0] = hiHalf ? d2lo : d0;
            A1[1] = hiHalf ? 0.0f : d1;

    // A fragment #2: rows = samples of lanes 16..31
    const float d0hi = __shfl(d0, lane | 16, 32);
    const float d1hi = __shfl(d1, lane | 16, 32);
    v2f A2; A2[0] = hiHalf ? d2 : d0hi;
            A2[1] = hiHalf ? 0.0f : d1hi;

    // (neg_a, A, neg_b, B, c_mod, C, reuse_a, reuse_b)
    v8f acc1 = __builtin_amdgcn_wmma_f32_16x16x4_f32(
        false, A1, false, Bm, (short)0, C, false, false);
    v8f acc2 = __builtin_amdgcn_wmma_f32_16x16x4_f32(
        false, A2, false, Bm, (short)0, C, false, false);

    // Redistribute C-layout (VGPR r, lane l: M=r+8*(l/16), N=l%16) back to
    // thread-per-sample: lane l owns sample m=l%16 of matrix (l<16 ? acc1 : acc2).
    const int rNeed   = lane & 7;
    const int srcBase = (lane & 8) ? 16 : 0;
    h0 = 0.0f; h1 = 0.0f;
#pragma unroll
    for (int r = 0; r < 8; ++r) {
        const float a0 = __shfl(acc1[r], srcBase,     32);
        const float a1 = __shfl(acc1[r], srcBase + 1, 32);
        const float c0 = __shfl(acc2[r], srcBase,     32);
        const float c1 = __shfl(acc2[r], srcBase + 1, 32);
        const float s0 = hiHalf ? c0 : a0;
        const float s1 = hiHalf ? c1 : a1;
        h0 = (rNeed == r) ? s0 : h0;
        h1 = (rNeed == r) ? s1 : h1;
    }
#else
    h0 = fmaf(d2, w1_02, fmaf(d1, w1_01, fmaf(d0, w1_00, bb1_0)));
    h1 = fmaf(d2, w1_12, fmaf(d1, w1_11, fmaf(d0, w1_10, bb1_1)));
#endif

    // ---- Phase 4: ELU + layers 2/3 in VALU --------------------------------
    const float e0 = eluf(h0), e1 = eluf(h1);
    const float g0 = fmaf(e1, w2_01, fmaf(e0, w2_00, bb2_0));
    const float g1 = fmaf(e1, w2_11, fmaf(e0, w2_10, bb2_1));
    const float f0 = eluf(g0), f1 = eluf(g1);
    const float y  = fmaf(f1, w3_1, fmaf(f0, w3_0, bb3));

    __builtin_nontemporal_store(y, out + (size_t)blockIdx.x * TPB + t);
}

extern "C" void kernel_launch(void* const* d_in, const int* in_sizes, int n_in,
                              void* d_out, int out_size, void* d_ws, size_t ws_size,
                              hipStream_t stream) {
    (void)n_in; (void)d_ws; (void)ws_size; (void)out_size;
    const float* X  = (const float*)d_in[0];
    const float* W1 = (const float*)d_in[1];
    const float* b1 = (const float*)d_in[2];
    const float* W2 = (const float*)d_in[3];
    const float* b2 = (const float*)d_in[4];
    const float* W3 = (const float*)d_in[5];
    const float* b3 = (const float*)d_in[6];
    float* out = (float*)d_out;

    const int B = in_sizes[0] / 9;      // 4,194,304 -> multiple of 256 (no tail)
    const int grid = B / TPB;           // 16384 blocks x 8 wave32

    pairdist_mlp_kernel<<<grid, TPB, 0, stream>>>(X, W1, b1, W2, b2, W3, b3, out);
}